// STMetaGCGRUCell_67920612819187
// MI455X (gfx1250) — compile-verified
//
#include <hip/hip_runtime.h>
#include <math.h>

#define CHEB_K 3
#define HIDDEN 64
#define INPUT_DIM 2
#define BB 16
#define NN 307
#define CIN (INPUT_DIM + HIDDEN)   // 66
#define INDIM (CHEB_K * CIN)       // 198
#define GATE2 (2 * HIDDEN)         // 128
#define NPOS (BB * NN)             // 4912

// padded dims (multiples of 16, K padded for uniform WMMA loop)
#define NP 320                     // padded N (rows & K dim)
#define CP 80                      // padded CIN
#define SP_ROW (CHEB_K * CP)       // 240: padded sup row

typedef __attribute__((ext_vector_type(2))) float v2f;
typedef __attribute__((ext_vector_type(8))) float v8f;

// ---------------------------------------------------------------------------
// zero-fill (graph-capture-safe memset), float4 stores
// ---------------------------------------------------------------------------
__global__ void k_zero(float4* __restrict__ p, int n4) {
    int idx = blockIdx.x * blockDim.x + threadIdx.x;
    if (idx < n4) p[idx] = make_float4(0.f, 0.f, 0.f, 0.f);
}

// ---------------------------------------------------------------------------
// Gp[k][n][m] (3,320,320) <- G[k][n][m] (3,307,307)
// ---------------------------------------------------------------------------
__global__ void k_fill_g(const float* __restrict__ G, float* __restrict__ Gp) {
    int idx = blockIdx.x * blockDim.x + threadIdx.x;
    const int total = CHEB_K * NN * NN;
    if (idx >= total) return;
    int m = idx % NN;
    int r = (idx / NN) % NN;
    int k = idx / (NN * NN);
    Gp[((size_t)k * NP + r) * NP + m] = G[idx];
}

// ---------------------------------------------------------------------------
// XpT[b][c][n] (B,80,320) <- combined[b][n][c]  (transposed, padded)
// idx ordered (b,c,n) so writes are coalesced over n.
// ---------------------------------------------------------------------------
__global__ void k_fill_x(const float* __restrict__ xt, const float* __restrict__ h,
                         float* __restrict__ XpT) {
    int idx = blockIdx.x * blockDim.x + threadIdx.x;
    const int total = BB * CIN * NN;
    if (idx >= total) return;
    int n = idx % NN;
    int c = (idx / NN) % CIN;
    int b = idx / (NN * CIN);
    int pos = b * NN + n;
    float v = (c < INPUT_DIM) ? xt[pos * INPUT_DIM + c]
                              : h[pos * HIDDEN + (c - INPUT_DIM)];
    XpT[((size_t)b * CP + c) * NP + n] = v;
}

// ---------------------------------------------------------------------------
// Sp[b][n][k*80+i] = sum_m Gp[k][n][m] * XpT[b][i][m]
// One wave per 16x16 tile. No bounds checks: everything padded.
// Inner loop: 2 x global_load_b64 + v_wmma_f32_16x16x4_f32, 80 iters.
// grid = (NP/16=20, CP/16=5, B*CHEB_K=48), block = 32.
// ---------------------------------------------------------------------------
__global__ void k_gcn_wmma(const float* __restrict__ Gp, const float* __restrict__ XpT,
                           float* __restrict__ Sp) {
    const int lane = threadIdx.x;      // 0..31
    const int half = lane >> 4;        // 0 | 1
    const int l    = lane & 15;

    const int ntile = blockIdx.x;
    const int itile = blockIdx.y;
    const int b = blockIdx.z / CHEB_K;
    const int k = blockIdx.z - b * CHEB_K;

    const int row = ntile * 16 + l;    // n
    const int col = itile * 16 + l;    // i
    // A 16x4 f32: VGPR0 = {K=0 | K=2}, VGPR1 = {K=1 | K=3} -> adjacent pair at 2*half
    const float* __restrict__ pa = Gp  + ((size_t)k * NP + row) * NP + 2 * half;
    const float* __restrict__ pb = XpT + ((size_t)b * CP + col) * NP + 2 * half;

    v8f acc = {};
#pragma unroll 4
    for (int m0 = 0; m0 < NP; m0 += 4) {
        v2f a  = *(const v2f*)(pa + m0);
        v2f bm = *(const v2f*)(pb + m0);
        acc = __builtin_amdgcn_wmma_f32_16x16x4_f32(
            false, a, false, bm, (short)0, acc, false, false);
    }
    // D: VGPR j -> row M = j + 8*half, col N = l
    float* __restrict__ ps =
        Sp + ((size_t)b * NP + ntile * 16 + 8 * half) * SP_ROW + k * CP + itile * 16 + l;
#pragma unroll
    for (int j = 0; j < 8; ++j)
        ps[(size_t)j * SP_ROW] = acc[j];
}

// ---------------------------------------------------------------------------
// gates = sigmoid(sup . W_gate + b_gate); z stored; candT = [xt, r*h] (transposed)
// One wave per position; float4 W loads -> 512B coalesced per row.
// Streams the ~498MB W_conv_gate at HBM rate (the real bottleneck).
// ---------------------------------------------------------------------------
__global__ void k_gates(const float* __restrict__ Sp, const float* __restrict__ W,
                        const float* __restrict__ bias, const float* __restrict__ xt,
                        const float* __restrict__ h, float* __restrict__ z,
                        float* __restrict__ candT) {
    const int pos = blockIdx.x;     // 0..NPOS-1
    const int t = threadIdx.x;      // 0..31
    const int b = pos / NN;
    const int n = pos - b * NN;

    __shared__ float s[INDIM];
    const float* __restrict__ spp = Sp + ((size_t)b * NP + n) * SP_ROW;
    for (int j = t; j < INDIM; j += 32)
        s[j] = spp[(j / CIN) * CP + (j % CIN)];
    __syncthreads();

    const float4* __restrict__ W4 = (const float4*)(W + (size_t)pos * INDIM * GATE2);
    float4 acc = ((const float4*)(bias + (size_t)pos * GATE2))[t];
#pragma unroll 2
    for (int i = 0; i < INDIM; ++i) {
        const float sv = s[i];
        const float4 w = W4[i * 32 + t];
        acc.x = fmaf(sv, w.x, acc.x);
        acc.y = fmaf(sv, w.y, acc.y);
        acc.z = fmaf(sv, w.z, acc.z);
        acc.w = fmaf(sv, w.w, acc.w);
    }
    float g[4] = {acc.x, acc.y, acc.z, acc.w};
#pragma unroll
    for (int u = 0; u < 4; ++u) {
        const int c = 4 * t + u;
        const float gv = 1.0f / (1.0f + __expf(-g[u]));
        if (c < HIDDEN) {
            z[(size_t)pos * HIDDEN + c] = gv;                        // z gate
        } else {
            const int cc = c - HIDDEN;                               // r gate
            candT[((size_t)b * CP + INPUT_DIM + cc) * NP + n] =
                gv * h[(size_t)pos * HIDDEN + cc];
        }
    }
    if (t < INPUT_DIM)
        candT[((size_t)b * CP + t) * NP + n] = xt[(size_t)pos * INPUT_DIM + t];
}

// ---------------------------------------------------------------------------
// n = tanh(sup2 . W_update + b_update); out = z*n + (1-z)*h
// One wave per position; float2 W loads -> 256B coalesced per row (~249MB).
// ---------------------------------------------------------------------------
__global__ void k_update(const float* __restrict__ Sp, const float* __restrict__ W,
                         const float* __restrict__ bias, const float* __restrict__ z,
                         const float* __restrict__ h, float* __restrict__ out) {
    const int pos = blockIdx.x;
    const int t = threadIdx.x;      // 0..31
    const int b = pos / NN;
    const int n = pos - b * NN;

    __shared__ float s[INDIM];
    const float* __restrict__ spp = Sp + ((size_t)b * NP + n) * SP_ROW;
    for (int j = t; j < INDIM; j += 32)
        s[j] = spp[(j / CIN) * CP + (j % CIN)];
    __syncthreads();

    const float2* __restrict__ W2 = (const float2*)(W + (size_t)pos * INDIM * HIDDEN);
    float2 acc = ((const float2*)(bias + (size_t)pos * HIDDEN))[t];
#pragma unroll 2
    for (int i = 0; i < INDIM; ++i) {
        const float sv = s[i];
        const float2 w = W2[i * 32 + t];
        acc.x = fmaf(sv, w.x, acc.x);
        acc.y = fmaf(sv, w.y, acc.y);
    }
#pragma unroll
    for (int u = 0; u < 2; ++u) {
        const int c = 2 * t + u;
        const float nv = tanhf(u == 0 ? acc.x : acc.y);
        const float zv = z[(size_t)pos * HIDDEN + c];
        const float hv = h[(size_t)pos * HIDDEN + c];
        out[(size_t)pos * HIDDEN + c] = zv * nv + (1.0f - zv) * hv;
    }
}

// ---------------------------------------------------------------------------
extern "C" void kernel_launch(void* const* d_in, const int* in_sizes, int n_in,
                              void* d_out, int out_size, void* d_ws, size_t ws_size,
                              hipStream_t stream) {
    const float* G      = (const float*)d_in[0];   // (3,307,307)
    const float* xt     = (const float*)d_in[1];   // (16,307,2)
    const float* h      = (const float*)d_in[2];   // (16,307,64)
    const float* W_gate = (const float*)d_in[3];   // (16,307,198,128)
    const float* b_gate = (const float*)d_in[4];   // (16,307,128)
    const float* W_upd  = (const float*)d_in[5];   // (16,307,198,64)
    const float* b_upd  = (const float*)d_in[6];   // (16,307,64)
    float* out = (float*)d_out;                    // (16,307,64)

    // workspace layout (floats): Gp | XpT (reused as candT) | Sp | z   (~9 MB)
    float* ws  = (float*)d_ws;
    float* Gp  = ws;                                       // 3*320*320   = 307200
    float* XpT = Gp  + (size_t)CHEB_K * NP * NP;           // 16*80*320   = 409600
    float* Spb = XpT + (size_t)BB * CP * NP;               // 16*320*240  = 1228800
    float* zb  = Spb + (size_t)BB * NP * SP_ROW;           // 16*307*64   = 314368

    const dim3 gcnGrid(NP / 16, CP / 16, BB * CHEB_K);     // (20,5,48)

    // zero padded buffers (padding must be 0 in the K dimension)
    {
        int n4g = (CHEB_K * NP * NP) / 4;
        k_zero<<<(n4g + 255) / 256, 256, 0, stream>>>((float4*)Gp, n4g);
        int n4x = (BB * CP * NP) / 4;
        k_zero<<<(n4x + 255) / 256, 256, 0, stream>>>((float4*)XpT, n4x);
    }
    // fill padded G and transposed combined = [xt, h]
    {
        int tg = CHEB_K * NN * NN;
        k_fill_g<<<(tg + 255) / 256, 256, 0, stream>>>(G, Gp);
        int tx = BB * CIN * NN;
        k_fill_x<<<(tx + 255) / 256, 256, 0, stream>>>(xt, h, XpT);
    }
    // sup1 = G (x) combined   (WMMA)
    k_gcn_wmma<<<gcnGrid, 32, 0, stream>>>(Gp, XpT, Spb);
    // gates -> z, candT (written back into XpT; padding stays zero)
    k_gates<<<NPOS, 32, 0, stream>>>(Spb, W_gate, b_gate, xt, h, zb, XpT);
    // sup2 = G (x) cand       (WMMA)
    k_gcn_wmma<<<gcnGrid, 32, 0, stream>>>(Gp, XpT, Spb);
    // update matvec + GRU blend -> out
    k_update<<<NPOS, 32, 0, stream>>>(Spb, W_upd, b_upd, zb, h, out);
}